// SequenceClassifier_34978213658858
// MI455X (gfx1250) — compile-verified
//
#include <hip/hip_runtime.h>
#include <hip/hip_bf16.h>
#include <math.h>

typedef __attribute__((ext_vector_type(2))) float v2f;
typedef __attribute__((ext_vector_type(8))) float v8f;

#define NN 32
#define EMBED 496
#define NCOMMS 50
#define LDAP 33   // padded LDS row stride (floats) to dodge bank conflicts

// R = P @ Q   (32x32 f32, row-major stride LDAP, all in LDS)
// Uses V_WMMA_F32_16X16X4_F32: A=16x4, B=4x16, C=16x16 f32.
// A frag: lanes 0-15 hold M=lane, v0=K(2*khalf), v1=K(2*khalf+1), khalf=lane>>4
// B frag: v0 = row (2*khalf), v1 = row (2*khalf+1), N = lane&15
// C frag: VGPR r = rows (r, r+8) of the 16-row tile, N = lane&15
__device__ __forceinline__ void mm32(const float* __restrict__ P,
                                     const float* __restrict__ Q,
                                     float* __restrict__ R, int lane) {
  const int lm = lane & 15;
  const int khalf = lane >> 4;           // 0 or 1
#pragma unroll
  for (int mt = 0; mt < 2; ++mt) {
#pragma unroll
    for (int nt = 0; nt < 2; ++nt) {
      v8f c = {};
#pragma unroll
      for (int k = 0; k < 8; ++k) {
        const int k0 = k * 4 + khalf * 2;
        v2f a, b;
        a.x = P[(mt * 16 + lm) * LDAP + k0];
        a.y = P[(mt * 16 + lm) * LDAP + k0 + 1];
        b.x = Q[(k0) * LDAP + nt * 16 + lm];
        b.y = Q[(k0 + 1) * LDAP + nt * 16 + lm];
        c = __builtin_amdgcn_wmma_f32_16x16x4_f32(
            /*neg_a=*/false, a, /*neg_b=*/false, b,
            /*c_mod=*/(short)0, c, /*reuse_a=*/false, /*reuse_b=*/false);
      }
      const int rrow = mt * 16 + khalf * 8;
      const int rcol = nt * 16 + lm;
#pragma unroll
      for (int r = 0; r < 8; ++r) R[(rrow + r) * LDAP + rcol] = c[r];
    }
  }
}

__global__ __launch_bounds__(32) void seqclass_expm_scan(
    const int* __restrict__ text,          // [S,B]
    const float* __restrict__ embed_table, // [N_TOKENS, 496]
    const float* __restrict__ Wmat,        // [50, 32]
    const float* __restrict__ bvec,        // [50]
    float* __restrict__ out,               // [B, 50]
    int S, int B) {
  __shared__ float xe[EMBED];
  __shared__ float Mb[4][NN * LDAP];  // 0: A (scaled), 1/2: Taylor term ping-pong, 3: E
  __shared__ float hbuf[NN];
  __shared__ float logits[NCOMMS];

  const int b = blockIdx.x;
  const int lane = threadIdx.x;      // 0..31, one wave
  const int i = lane;                // this lane owns matrix row i

  hbuf[lane] = (lane == 0) ? 1.0f : 0.0f;

  for (int s = 0; s < S; ++s) {
    // ---- gather embedding row for this token into LDS ----
    const int tok = text[(size_t)s * B + b];
    const float* row = embed_table + (size_t)tok * EMBED;
#pragma unroll 4
    for (int k = lane; k < EMBED; k += 32) xe[k] = row[k];

    // prefetch next step's embedding row (16 lanes x 128B covers 1984B)
    if (s + 1 < S) {
      const int tok2 = text[(size_t)(s + 1) * B + b];
      const float* row2 = embed_table + (size_t)tok2 * EMBED;
      if (lane < 16) __builtin_prefetch(row2 + lane * 32, 0, 1);
    }
    __syncthreads();

    // ---- build skew-symmetric A: A[i][j] = x[ix[i,j]] - x[ix[j,i]] ----
    float* A = Mb[0];
#pragma unroll
    for (int j = 0; j < NN; ++j) {
      float v = 0.0f;
      if (j > i)      v =  xe[i * (61 - i) / 2 + j - 1];
      else if (j < i) v = -xe[j * (61 - j) / 2 + i - 1];
      A[i * LDAP + j] = v;
    }
    __syncthreads();

    // ---- 1-norm (max column abs-sum), wave reduce ----
    float cs = 0.0f;
#pragma unroll
    for (int r = 0; r < NN; ++r) cs += fabsf(A[r * LDAP + lane]);
#pragma unroll
    for (int off = 16; off > 0; off >>= 1) cs = fmaxf(cs, __shfl_xor(cs, off, 32));

    // ---- scaling: B = A / 2^sq with ||B||_1 <= 0.25 ----
    int sq = 0;
    float nrm = cs;
    while (nrm > 0.25f && sq < 24) { nrm *= 0.5f; ++sq; }
    const float sc = ldexpf(1.0f, -sq);
#pragma unroll
    for (int j = 0; j < NN; ++j) A[i * LDAP + j] *= sc;
    __syncthreads();

    // ---- Taylor: E = I + B + B^2/2! + ... + B^9/9! ----
    float* T   = Mb[1];
    float* Tn  = Mb[2];
    float* E   = Mb[3];
#pragma unroll
    for (int j = 0; j < NN; ++j) {
      const float av = A[i * LDAP + j];
      T[i * LDAP + j] = av;
      E[i * LDAP + j] = av + ((i == j) ? 1.0f : 0.0f);
    }
    __syncthreads();

    for (int k = 2; k <= 9; ++k) {
      mm32(A, T, Tn, lane);              // Tn = A @ T_{k-1}
      __syncthreads();
      const float rk = 1.0f / (float)k;  // T_k = Tn / k
#pragma unroll
      for (int j = 0; j < NN; ++j) {
        const float tv = Tn[i * LDAP + j] * rk;
        Tn[i * LDAP + j] = tv;
        E[i * LDAP + j] += tv;
      }
      __syncthreads();
      float* tmp = T; T = Tn; Tn = tmp;  // ping-pong
    }

    // ---- squaring: E <- E^2, sq times ----
    for (int t = 0; t < sq; ++t) {
      mm32(E, E, Tn, lane);
      __syncthreads();
      float* tmp = E; E = Tn; Tn = tmp;
    }

    // ---- recurrence: h_new[j] = sum_i E[i][j] * h[i] ----
    float acc = 0.0f;
#pragma unroll
    for (int r = 0; r < NN; ++r) acc = fmaf(E[r * LDAP + lane], hbuf[r], acc);
    __syncthreads();
    hbuf[lane] = acc;
    __syncthreads();
  }

  // ---- logits = h @ W^T + b ----
  for (int c = lane; c < NCOMMS; c += 32) {
    float acc = bvec[c];
#pragma unroll
    for (int j = 0; j < NN; ++j) acc = fmaf(Wmat[c * NN + j], hbuf[j], acc);
    logits[c] = acc;
  }
  __syncthreads();

  // ---- softmax over 50 entries (wave-cooperative) ----
  float mx = -INFINITY;
  for (int c = lane; c < NCOMMS; c += 32) mx = fmaxf(mx, logits[c]);
#pragma unroll
  for (int off = 16; off > 0; off >>= 1) mx = fmaxf(mx, __shfl_xor(mx, off, 32));
  float sum = 0.0f;
  for (int c = lane; c < NCOMMS; c += 32) sum += __expf(logits[c] - mx);
#pragma unroll
  for (int off = 16; off > 0; off >>= 1) sum += __shfl_xor(sum, off, 32);
  const float rs = 1.0f / sum;
  for (int c = lane; c < NCOMMS; c += 32)
    out[(size_t)b * NCOMMS + c] = __expf(logits[c] - mx) * rs;
}

extern "C" void kernel_launch(void* const* d_in, const int* in_sizes, int n_in,
                              void* d_out, int out_size, void* d_ws, size_t ws_size,
                              hipStream_t stream) {
  const int*   text        = (const int*)d_in[0];   // [S,B]
  // d_in[1] = text_lens (unused by reference; used only to recover B)
  const float* embed_table = (const float*)d_in[2]; // [N_TOKENS, 496]
  const float* Wmat        = (const float*)d_in[3]; // [50, 32]
  const float* bvec        = (const float*)d_in[4]; // [50]
  float* out = (float*)d_out;

  const int B = in_sizes[1];            // batch = len(text_lens)
  const int S = in_sizes[0] / B;        // seq len

  seqclass_expm_scan<<<B, 32, 0, stream>>>(text, embed_table, Wmat, bvec, out, S, B);
}